// IntSoftmax_43774306680979
// MI455X (gfx1250) — compile-verified
//
#include <hip/hip_runtime.h>

// ---------------- IntSoftmax (I-BERT) constants ----------------
#define NCOLS   2048
#define TPB     256
#define VPT     8            // NCOLS / TPB
#define COEF0_F 0.35815147f
#define COEF1_F (0.96963238f / 0.35815147f)
#define COEF2_F (1.0f / 0.35815147f)
#define X0_F    (-0.6931f)
#define CONSTQ  30

// CDNA5 async global->LDS path (gfx1250). Guarded so compile never fails;
// fallback is a plain register-staged copy.
#if defined(__gfx1250__) && __has_builtin(__builtin_amdgcn_global_load_async_to_lds_b128)
#define HAS_ASYNC_LDS 1
#else
#define HAS_ASYNC_LDS 0
#endif

__device__ __forceinline__ float wave_red_max(float v) {
#pragma unroll
  for (int off = 16; off > 0; off >>= 1)
    v = fmaxf(v, __shfl_xor(v, off, 32));
  return v;
}
__device__ __forceinline__ float wave_red_sum(float v) {
#pragma unroll
  for (int off = 16; off > 0; off >>= 1)
    v += __shfl_xor(v, off, 32);
  return v;
}

__global__ __launch_bounds__(TPB) void intsoftmax_kernel(
    const float* __restrict__ x, const float* __restrict__ sf_p,
    float* __restrict__ out, int rows) {
  __shared__ float lds_row[NCOLS];          // 8 KB row tile
  __shared__ float red[TPB / 32];
  __shared__ float bcast;

  const int row = blockIdx.x;
  if (row >= rows) return;
  const int t = threadIdx.x;
  const size_t base = (size_t)row * NCOLS + (size_t)t * VPT;

  const float sf     = sf_p[0];
  const float inv_sf = 1.0f / sf;                         // hoisted: 1 div/thread
  const float x0_int = floorf(X0_F * inv_sf);             // negative (-178 @ sf=1/256)
  const float inv_x0 = 1.0f / x0_int;                     // hoisted: 1 div/thread
  const float b_int  = floorf(COEF1_F * inv_sf);
  const float c_int  = floorf(COEF2_F * (inv_sf * inv_sf));
  const float qmin   = (float)CONSTQ * x0_int;
  // Global max of exp_int is attained at x_int==0 (present in every row):
  // exp_int_max = floor(c_int * 2^30). exp_sf cancels in qv = x_real/act_sf.
  const float emax    = floorf(ldexpf(c_int, CONSTQ));
  const float requant = 32767.0f / emax;

  // ---- stage this row into LDS via CDNA5 async global->LDS DMA ----
  const float* g = x + base;
#if HAS_ASYNC_LDS
  {
    typedef int v4i __attribute__((ext_vector_type(4)));
    typedef __attribute__((address_space(1))) v4i* gvp;   // global (AS1)
    typedef __attribute__((address_space(3))) v4i* lvp;   // LDS (AS3)
    gvp g0 = (gvp)(g);
    gvp g1 = (gvp)(g + 4);
    lvp l0 = (lvp)(&lds_row[t * VPT]);
    lvp l1 = (lvp)(&lds_row[t * VPT + 4]);
    __builtin_amdgcn_global_load_async_to_lds_b128(g0, l0, 0, 0);
    __builtin_amdgcn_global_load_async_to_lds_b128(g1, l1, 0, 0);
#if __has_builtin(__builtin_amdgcn_s_wait_asynccnt)
    __builtin_amdgcn_s_wait_asynccnt(0);
#else
    asm volatile("s_wait_asynccnt 0" ::: "memory");
#endif
  }
#else
  {
    const float4* gv = (const float4*)g;
    float4 a = gv[0];
    float4 b = gv[1];
    *(float4*)&lds_row[t * VPT]     = a;
    *(float4*)&lds_row[t * VPT + 4] = b;
  }
#endif
  __syncthreads();

  // ---- pull row slice to registers (ds_load_b128), compute x_int & row max ----
  float v[VPT];
  {
    float4 a = *(const float4*)&lds_row[t * VPT];
    float4 b = *(const float4*)&lds_row[t * VPT + 4];
    v[0] = a.x; v[1] = a.y; v[2] = a.z; v[3] = a.w;
    v[4] = b.x; v[5] = b.y; v[6] = b.z; v[7] = b.w;
  }
  float m = -__builtin_inff();
#pragma unroll
  for (int i = 0; i < VPT; ++i) {
    v[i] = v[i] * inv_sf;                // x_int (mul, not full IEEE div)
    m = fmaxf(m, v[i]);
  }
  // block-wide max: wave shfl reduce, then 8 partials through LDS
  m = wave_red_max(m);
  if ((t & 31) == 0) red[t >> 5] = m;
  __syncthreads();
  if (t < 32) {
    float r = (t < TPB / 32) ? red[t] : -__builtin_inff();
    r = wave_red_max(r);
    if (t == 0) bcast = r;
  }
  __syncthreads();
  const float rowmax = bcast;

  // ---- int_exp + int_polynomial + QuantAct(16) requant, running row sum ----
  float qv2[VPT];
  float s = 0.0f;
#pragma unroll
  for (int i = 0; i < VPT; ++i) {
    float xi = v[i] - rowmax;                       // <= 0
    xi = fmaxf(xi, qmin);
    float q = floorf(xi * inv_x0);                  // 0..30 (mul, not div)
    float r = xi - x0_int * q;
    float z = (r + b_int) * r + c_int;
    float e = floorf(ldexpf(z, CONSTQ - (int)q));   // z * 2^(30-q), exact scale
    e = fmaxf(e, 0.0f);
    float qq = rintf(e * requant);                  // round-to-nearest-even
    qq = fminf(fmaxf(qq, -32768.0f), 32767.0f);
    qv2[i] = qq;
    s += qq;
  }
  // block-wide sum
  s = wave_red_sum(s);
  if ((t & 31) == 0) red[t >> 5] = s;
  __syncthreads();
  if (t < 32) {
    float r = (t < TPB / 32) ? red[t] : 0.0f;
    r = wave_red_sum(r);
    if (t == 0) bcast = r;
  }
  __syncthreads();
  const float rowsum = bcast;

  // ---- normalize via bit-shift, write out (two b128 stores) ----
  const float factor = floorf(4294967296.0f / rowsum);    // once per thread
  const float fscale = factor * (1.0f / 16777216.0f);     // exact pow2 scale
#pragma unroll
  for (int i = 0; i < VPT; ++i)
    qv2[i] = floorf(qv2[i] * fscale) * 0.00390625f;       // == floor(q*f/2^24)*2^-8

  float4 o0 = make_float4(qv2[0], qv2[1], qv2[2], qv2[3]);
  float4 o1 = make_float4(qv2[4], qv2[5], qv2[6], qv2[7]);
  *(float4*)(out + base)     = o0;
  *(float4*)(out + base + 4) = o1;
}

extern "C" void kernel_launch(void* const* d_in, const int* in_sizes, int n_in,
                              void* d_out, int out_size, void* d_ws, size_t ws_size,
                              hipStream_t stream) {
  (void)n_in; (void)out_size; (void)d_ws; (void)ws_size;
  const float* x  = (const float*)d_in[0];
  const float* sf = (const float*)d_in[1];
  float* out      = (float*)d_out;
  const int rows  = in_sizes[0] / NCOLS;   // 16 * 2048 = 32768 rows
  intsoftmax_kernel<<<dim3(rows), dim3(TPB), 0, stream>>>(x, sf, out, rows);
}